// CopyNet_2516850835869
// MI455X (gfx1250) — compile-verified
//
#include <hip/hip_runtime.h>

#define NB 4
#define NS 512
#define NT 512
#define ND 256

typedef __attribute__((ext_vector_type(2))) float v2f;
typedef __attribute__((ext_vector_type(8))) float v8f;

__device__ __forceinline__ float fast_tanh(float x) {
#if __has_builtin(__builtin_amdgcn_tanhf)
    return __builtin_amdgcn_tanhf(x);      // v_tanh_f32 (gfx1250 transcendental)
#else
    // tanh(x) = 1 - 2/(e^{2x}+1), via v_exp_f32 + v_rcp_f32
    float z = fminf(fmaxf(x, -20.0f), 20.0f);
    float t = __builtin_amdgcn_exp2f(z * 2.885390081777927f); // 2*log2(e)
    return 1.0f - 2.0f * __builtin_amdgcn_rcpf(t + 1.0f);
#endif
}

// Y = X @ W + bias, X: [NB][R][ND] row-major, W: [ND][ND] row-major.
// TRANSPOSE==1: store Y as Yt[b][d][row] (leading dim R)   (for src path)
// TRANSPOSE==0: store Y as Y [b][row][d]                   (for tgt path)
// One wave (32 lanes) per 16x16 output tile, K-loop in steps of 4 using
// V_WMMA_F32_16X16X4_F32.
template <int TRANSPOSE, int R>
__global__ void proj_wmma(const float* __restrict__ X,
                          const float* __restrict__ W,
                          const float* __restrict__ bias,
                          float* __restrict__ Y) {
    const int tile = blockIdx.x;                 // NB * (R/16) * (ND/16) tiles
    const int dt   = tile % (ND / 16);
    const int rt   = (tile / (ND / 16)) % (R / 16);
    const int b    = tile / ((ND / 16) * (R / 16));
    const int d0   = dt * 16;
    const int r0   = rt * 16;

    const int lane = threadIdx.x & 31;
    const int m    = lane & 15;                  // M (A row / D col) index
    const int half = lane >> 4;                  // 0 or 1

    const float* Xb = X + (size_t)b * R * ND + (size_t)(r0 + m) * ND;
    v8f acc = {};

    for (int k = 0; k < ND; k += 4) {
        const int klo = k + 2 * half;
        // A fragment: A[M = r0+m, K = klo..klo+1] -> consecutive in memory
        v2f a;
        a.x = Xb[klo];
        a.y = Xb[klo + 1];
        // B fragment: B[K = klo..klo+1, N = d0+m]
        v2f bm;
        bm.x = W[(size_t)klo * ND + d0 + m];
        bm.y = W[(size_t)(klo + 1) * ND + d0 + m];
        acc = __builtin_amdgcn_wmma_f32_16x16x4_f32(
            false, a, false, bm, (short)0, acc, false, false);
    }

    const float bv = bias[d0 + m];
#pragma unroll
    for (int v = 0; v < 8; ++v) {
        const int row = r0 + v + 8 * half;       // C/D layout: M = v + 8*half
        const float val = acc[v] + bv;
        if (TRANSPOSE)
            Y[(size_t)b * ND * R + (size_t)(d0 + m) * R + row] = val;
        else
            Y[(size_t)b * R * ND + (size_t)row * ND + d0 + m] = val;
    }
}

// Fused Bahdanau score: genP[b,t,s] = sum_d tanh(srcT[b,d,s] + tgt[b,t,d]) * Wres[d] + br
// One block = one b, TCHUNK consecutive t, all S=512 s (2 per thread).
#define TCHUNK 4
__global__ void __launch_bounds__(256)
genp_kernel(const float* __restrict__ srcT,   // [NB][ND][NS]
            const float* __restrict__ tgtL,   // [NB][NT][ND]
            const float* __restrict__ Wres,   // [ND]
            const float* __restrict__ bres,   // [1]
            float* __restrict__ genP) {       // [NB][NT][NS]
    __shared__ float s_tgt[TCHUNK][ND];
    __shared__ float s_w[ND];

    const int blk   = blockIdx.x;                    // NB*NT/TCHUNK blocks
    const int b     = blk / (NT / TCHUNK);
    const int tbase = (blk % (NT / TCHUNK)) * TCHUNK;
    const int tid   = threadIdx.x;                   // 256 threads

    s_w[tid] = Wres[tid];
#pragma unroll
    for (int c = 0; c < TCHUNK; ++c)
        s_tgt[c][tid] = tgtL[((size_t)b * NT + tbase + c) * ND + tid];
    __syncthreads();

    const float* sp = srcT + (size_t)b * ND * NS;
    float acc0[TCHUNK] = {};
    float acc1[TCHUNK] = {};

#pragma unroll 2
    for (int d = 0; d < ND; ++d) {
        const float w  = s_w[d];
        const float v0 = sp[(size_t)d * NS + tid];
        const float v1 = sp[(size_t)d * NS + tid + 256];
#pragma unroll
        for (int c = 0; c < TCHUNK; ++c) {
            const float tg = s_tgt[c][d];
            acc0[c] += fast_tanh(v0 + tg) * w;
            acc1[c] += fast_tanh(v1 + tg) * w;
        }
    }

    const float br = bres[0];
#pragma unroll
    for (int c = 0; c < TCHUNK; ++c) {
        const size_t row = (size_t)b * NT + tbase + c;
        genP[row * NS + tid]       = acc0[c] + br;
        genP[row * NS + tid + 256] = acc1[c] + br;
    }
}

// prob[b,t,:] = softmax(target[b,t,:] @ W_prob + b_prob) ; one wave per (b,t) row.
__global__ void __launch_bounds__(256)
prob_kernel(const float* __restrict__ target,  // [NB][NT][ND]
            const float* __restrict__ Wp,      // [ND][2]
            const float* __restrict__ bp,      // [2]
            float* __restrict__ prob) {        // [NB][NT][2]
    const int gtid = blockIdx.x * blockDim.x + threadIdx.x;
    const int row  = gtid >> 5;                   // global wave id = (b*NT + t)
    const int lane = gtid & 31;

    const float* xr = target + (size_t)row * ND;
    float a0 = 0.0f, a1 = 0.0f;
    for (int d = lane; d < ND; d += 32) {
        const float x = xr[d];
        a0 += x * Wp[2 * d + 0];
        a1 += x * Wp[2 * d + 1];
    }
#pragma unroll
    for (int off = 16; off > 0; off >>= 1) {
        a0 += __shfl_xor(a0, off, 32);
        a1 += __shfl_xor(a1, off, 32);
    }
    if (lane == 0) {
        const float e0 = a0 + bp[0];
        const float e1 = a1 + bp[1];
        const float mx = fmaxf(e0, e1);
        const float p0 = __expf(e0 - mx);
        const float p1 = __expf(e1 - mx);
        const float inv = 1.0f / (p0 + p1);
        prob[(size_t)row * 2 + 0] = p0 * inv;
        prob[(size_t)row * 2 + 1] = p1 * inv;
    }
}

extern "C" void kernel_launch(void* const* d_in, const int* in_sizes, int n_in,
                              void* d_out, int out_size, void* d_ws, size_t ws_size,
                              hipStream_t stream) {
    const float* source = (const float*)d_in[0];   // (B,S,D)
    const float* target = (const float*)d_in[1];   // (B,T,D)
    const float* W_src  = (const float*)d_in[2];   // (D,D)
    const float* b_src  = (const float*)d_in[3];   // (D,)
    const float* W_tgt  = (const float*)d_in[4];   // (D,D)
    const float* b_tgt  = (const float*)d_in[5];   // (D,)
    const float* W_res  = (const float*)d_in[6];   // (D,1)
    const float* b_res  = (const float*)d_in[7];   // (1,)
    const float* W_prob = (const float*)d_in[8];   // (D,2)
    const float* b_prob = (const float*)d_in[9];   // (2,)

    float* srcT = (float*)d_ws;                    // [NB][ND][NS] : 524288 f32
    float* tgtL = srcT + (size_t)NB * ND * NS;     // [NB][NT][ND] : 524288 f32

    float* genP = (float*)d_out;                               // B*T*S
    float* prob = genP + (size_t)NB * NT * NS;                 // B*T*2

    // Projections via fp32 WMMA (16x16x4), one wave per 16x16 tile.
    const int src_tiles = NB * (NS / 16) * (ND / 16);          // 2048
    const int tgt_tiles = NB * (NT / 16) * (ND / 16);          // 2048
    proj_wmma<1, NS><<<src_tiles, 32, 0, stream>>>(source, W_src, b_src, srcT);
    proj_wmma<0, NT><<<tgt_tiles, 32, 0, stream>>>(target, W_tgt, b_tgt, tgtL);

    // Fused tanh-attention score.
    genp_kernel<<<NB * NT / TCHUNK, 256, 0, stream>>>(srcT, tgtL, W_res, b_res, genP);

    // Tiny softmax head.
    prob_kernel<<<(NB * NT) / 8, 256, 0, stream>>>(target, W_prob, b_prob, prob);
}